// AttentionHead_38259568673459
// MI455X (gfx1250) — compile-verified
//
#include <hip/hip_runtime.h>
#include <math.h>

typedef float v2f __attribute__((ext_vector_type(2)));
typedef float v8f __attribute__((ext_vector_type(8)));

#define D_MODEL 1024
#define D_HID   64
#define NCTX    4096
#define NBATCH  4
#define TOTROWS (NBATCH * NCTX)        // 16384
#define NQT     (NBATCH * (NCTX / 16)) // 1024 query tiles (batch folded in)
#define NEG_BIG (-3.0e38f)

// D = A(16x4,f32) * B(4x16,f32) + C(16x16,f32), wave32.
__device__ __forceinline__ v8f wmma_f32(v2f a, v2f b, v8f c) {
  return __builtin_amdgcn_wmma_f32_16x16x4_f32(
      /*neg_a=*/false, a, /*neg_b=*/false, b,
      /*c_mod=*/(short)0, c, /*reuse_a=*/false, /*reuse_b=*/false);
}

// Fragment loader for A (16x4, M across lanes) and for B when the source is the
// row-major transpose ([N][K]): lane&15 selects the row, columns are
// kk + (lane>>4)*2 , +1.  Matches ISA 7.12.2 32-bit layouts.
__device__ __forceinline__ v2f frag_rm(const float* __restrict__ p, int ld,
                                       int row0, int kk, int lane) {
  const float* q = p + (size_t)(row0 + (lane & 15)) * ld + (kk + ((lane >> 4) << 1));
  v2f r = {q[0], q[1]};
  return r;
}

// ---------------------------------------------------------------------------
// Kernel 1: fused QKV projection, 2x2 register-blocked.
// Each wave: 32x32 output tile of one of Q/K/V (4 accumulators; 4 loads feed
// 4 WMMAs per k-chunk).  6 col-pair tiles (3 mats x 2) x 512 row tiles.
// ---------------------------------------------------------------------------
__global__ void qkv_kernel(const float* __restrict__ x,
                           const float* __restrict__ Wq, const float* __restrict__ bq,
                           const float* __restrict__ Wk, const float* __restrict__ bk,
                           const float* __restrict__ Wov, const float* __restrict__ bov,
                           float* __restrict__ Qo, float* __restrict__ Ko,
                           float* __restrict__ Vo) {
  const int lane = threadIdx.x & 31;
  const int wid  = (int)((blockIdx.x * blockDim.x + threadIdx.x) >> 5);
  const int wt   = wid % 6;
  const int rt   = wid / 6;
  if (rt >= TOTROWS / 32) return;
  const int mat = wt >> 1;             // 0=Q, 1=K, 2=V
  const int h0  = (wt & 1) << 5;       // 0 or 32
  const float* W    = (mat == 0) ? Wq : (mat == 1) ? Wk : Wov;
  const float* bias = (mat == 0) ? bq : (mat == 1) ? bk : bov;
  float*       Out  = (mat == 0) ? Qo : (mat == 1) ? Ko : Vo;
  const int m0 = rt << 5;

  v8f c00 = {}, c01 = {}, c10 = {}, c11 = {};
  for (int kk = 0; kk < D_MODEL; kk += 4) {
    v2f a0 = frag_rm(x, D_MODEL, m0,      kk, lane);
    v2f a1 = frag_rm(x, D_MODEL, m0 + 16, kk, lane);
    v2f b0 = frag_rm(W, D_MODEL, h0,      kk, lane);
    v2f b1 = frag_rm(W, D_MODEL, h0 + 16, kk, lane);
    c00 = wmma_f32(a0, b0, c00);
    c01 = wmma_f32(a0, b1, c01);
    c10 = wmma_f32(a1, b0, c10);
    c11 = wmma_f32(a1, b1, c11);
  }
  const int n  = lane & 15;
  const int hi = lane >> 4;
  const float bv0 = bias[h0 + n];
  const float bv1 = bias[h0 + 16 + n];
#pragma unroll
  for (int r = 0; r < 8; ++r) {
    const size_t r0 = (size_t)(m0 + r + 8 * hi) * D_HID;
    const size_t r1 = (size_t)(m0 + 16 + r + 8 * hi) * D_HID;
    Out[r0 + h0 + n]      = c00[r] + bv0;
    Out[r0 + h0 + 16 + n] = c01[r] + bv1;
    Out[r1 + h0 + n]      = c10[r] + bv0;
    Out[r1 + h0 + 16 + n] = c11[r] + bv1;
  }
}

// ---------------------------------------------------------------------------
// Kernel 2: causal flash attention, key-split into P partitions per q-tile.
// One wave per (q-tile, partition).  Writes UN-normalized partials
// (O_p, m_p, l_p); combine_kernel merges them.
// ---------------------------------------------------------------------------
__global__ void attn_kernel(const float* __restrict__ Q,
                            const float* __restrict__ K,
                            const float* __restrict__ V,
                            float* __restrict__ Opar,
                            float* __restrict__ Mpar,
                            float* __restrict__ Lpar,
                            int P) {
  __shared__ float ldsP[4][16 * 16];    // per-wave P staging (C-layout -> A-layout)
  const int lane = threadIdx.x & 31;
  const int lw   = threadIdx.x >> 5;
  const int wid  = (int)(blockIdx.x * (blockDim.x >> 5) + lw);
  const int p    = wid % P;
  const int qw   = wid / P;
  const int qt   = qw & 255;            // 256 query tiles per batch
  const int b    = qw >> 8;
  if (b >= NBATCH) return;
  const float* Qb = Q + (size_t)b * NCTX * D_HID;
  const float* Kb = K + (size_t)b * NCTX * D_HID;
  const float* Vb = V + (size_t)b * NCTX * D_HID;
  const int q0 = qt << 4;
  const int n  = lane & 15;
  const int hi = lane >> 4;
  float* myP = ldsP[lw];

  const int c   = qt + 1;               // causal key-tile count for this q-tile
  const int kt0 = (p * c) / P;
  const int kt1 = ((p + 1) * c) / P;

  // Q fragments in registers for the whole key loop; fold in 1/sqrt(d_model).
  const float scale = 0.03125f;  // 1/32
  v2f qf[16];
#pragma unroll
  for (int kk4 = 0; kk4 < 16; ++kk4) {
    v2f t = frag_rm(Qb, D_HID, q0, kk4 * 4, lane);
    v2f s = {t.x * scale, t.y * scale};
    qf[kk4] = s;
  }

  v8f o0 = {}, o1 = {}, o2 = {}, o3 = {};
  float mrow[8], lrow[8];
#pragma unroll
  for (int r = 0; r < 8; ++r) { mrow[r] = NEG_BIG; lrow[r] = 0.0f; }

  for (int kt = kt0; kt < kt1; ++kt) {
    const int k0 = kt << 4;
    // S = (Q*scale) @ K^T   (K-dim = 64 -> 16 chained WMMAs)
    v8f s = {};
#pragma unroll
    for (int kk4 = 0; kk4 < 16; ++kk4) {
      v2f bf = frag_rm(Kb, D_HID, k0, kk4 * 4, lane);   // B = K^T
      s = wmma_f32(qf[kk4], bf, s);
    }
    if (kt == qt) {   // causal mask on the diagonal tile: key N > query M
#pragma unroll
      for (int r = 0; r < 8; ++r) {
        const int M = r + 8 * hi;
        if (n > M) s[r] = NEG_BIG;
      }
    }
    // Online softmax: rowwise max/sum across the 16-lane half-groups.
#pragma unroll
    for (int r = 0; r < 8; ++r) {
      float rv = s[r];
      rv = fmaxf(rv, __shfl_xor(rv, 1, 32));
      rv = fmaxf(rv, __shfl_xor(rv, 2, 32));
      rv = fmaxf(rv, __shfl_xor(rv, 4, 32));
      rv = fmaxf(rv, __shfl_xor(rv, 8, 32));
      const float mn    = fmaxf(mrow[r], rv);
      const float alpha = __expf(mrow[r] - mn);
      mrow[r] = mn;
      const float pv = __expf(s[r] - mn);
      myP[(r + 8 * hi) * 16 + n] = pv;     // stage P row-major in LDS
      float rs = pv;
      rs += __shfl_xor(rs, 1, 32);
      rs += __shfl_xor(rs, 2, 32);
      rs += __shfl_xor(rs, 4, 32);
      rs += __shfl_xor(rs, 8, 32);
      lrow[r] = lrow[r] * alpha + rs;
      o0[r] *= alpha; o1[r] *= alpha; o2[r] *= alpha; o3[r] *= alpha;
    }
    // O += P @ V_tile.  P re-read from LDS in A-layout (same-wave DS ordering).
#pragma unroll
    for (int kk4 = 0; kk4 < 4; ++kk4) {
      const float* pp = myP + n * 16 + kk4 * 4 + hi * 2;
      v2f pa = {pp[0], pp[1]};
      const float* v0 = Vb + (size_t)(k0 + kk4 * 4 + hi * 2) * D_HID;
      const float* v1 = v0 + D_HID;
      v2f b0 = {v0[ 0 + n], v1[ 0 + n]};  o0 = wmma_f32(pa, b0, o0);
      v2f b1 = {v0[16 + n], v1[16 + n]};  o1 = wmma_f32(pa, b1, o1);
      v2f b2 = {v0[32 + n], v1[32 + n]};  o2 = wmma_f32(pa, b2, o2);
      v2f b3 = {v0[48 + n], v1[48 + n]};  o3 = wmma_f32(pa, b3, o3);
    }
  }
  // Store UN-normalized partial tile + per-row (m, l).
  const int part = (b * 256 + qt) * P + p;
  float* Od = Opar + (size_t)part * (16 * D_HID);
#pragma unroll
  for (int r = 0; r < 8; ++r) {
    const size_t row = (size_t)(r + 8 * hi) * D_HID;
    Od[row +  0 + n] = o0[r];
    Od[row + 16 + n] = o1[r];
    Od[row + 32 + n] = o2[r];
    Od[row + 48 + n] = o3[r];
  }
  if (n == 0) {      // lanes 0 and 16 cover rows 0..7 and 8..15
#pragma unroll
    for (int r = 0; r < 8; ++r) {
      Mpar[part * 16 + r + 8 * hi] = mrow[r];
      Lpar[part * 16 + r + 8 * hi] = lrow[r];
    }
  }
}

// ---------------------------------------------------------------------------
// Kernel 2b: merge partition partials into normalized O.
// grid = NQT blocks x 256 threads; each thread: one row, 4 consecutive cols.
// ---------------------------------------------------------------------------
__global__ void combine_kernel(const float* __restrict__ Opar,
                               const float* __restrict__ Mpar,
                               const float* __restrict__ Lpar,
                               float* __restrict__ O, int P) {
  const int qtile = blockIdx.x;               // 0..NQT-1 (batch folded in)
  const int t     = threadIdx.x;              // 0..255
  const int row   = t >> 4;                   // 0..15
  const int c0    = (t & 15) << 2;            // 0,4,...,60
  float mstar = NEG_BIG;
  for (int p = 0; p < P; ++p)
    mstar = fmaxf(mstar, Mpar[(qtile * P + p) * 16 + row]);
  float lstar = 0.0f, a0 = 0.0f, a1 = 0.0f, a2 = 0.0f, a3 = 0.0f;
  for (int p = 0; p < P; ++p) {
    const int part = qtile * P + p;
    const float w = __expf(Mpar[part * 16 + row] - mstar);
    lstar += w * Lpar[part * 16 + row];
    const float* src = Opar + (size_t)part * (16 * D_HID) + (size_t)row * D_HID + c0;
    a0 += w * src[0]; a1 += w * src[1]; a2 += w * src[2]; a3 += w * src[3];
  }
  const float inv = 1.0f / lstar;
  float* dst = O + (size_t)(qtile * 16 + row) * D_HID + c0;
  dst[0] = a0 * inv; dst[1] = a1 * inv; dst[2] = a2 * inv; dst[3] = a3 * inv;
}

// ---------------------------------------------------------------------------
// Kernel 3: output projection, 2x2 register-blocked.
// out[m,d] = sum_h O[m,h]*Wo[d,h] + bo[d].  32x32 tile per wave.
// ---------------------------------------------------------------------------
__global__ void proj_kernel(const float* __restrict__ O,
                            const float* __restrict__ Wo,
                            const float* __restrict__ bo,
                            float* __restrict__ out) {
  const int lane = threadIdx.x & 31;
  const int wid  = (int)((blockIdx.x * blockDim.x + threadIdx.x) >> 5);
  const int dt   = wid & 31;          // 32 column tiles of 32
  const int rt   = wid >> 5;
  if (rt >= TOTROWS / 32) return;
  const int m0 = rt << 5;
  const int d0 = dt << 5;
  v8f c00 = {}, c01 = {}, c10 = {}, c11 = {};
#pragma unroll
  for (int kk4 = 0; kk4 < 16; ++kk4) {
    v2f a0 = frag_rm(O,  D_HID, m0,      kk4 * 4, lane);
    v2f a1 = frag_rm(O,  D_HID, m0 + 16, kk4 * 4, lane);
    v2f b0 = frag_rm(Wo, D_HID, d0,      kk4 * 4, lane);
    v2f b1 = frag_rm(Wo, D_HID, d0 + 16, kk4 * 4, lane);
    c00 = wmma_f32(a0, b0, c00);
    c01 = wmma_f32(a0, b1, c01);
    c10 = wmma_f32(a1, b0, c10);
    c11 = wmma_f32(a1, b1, c11);
  }
  const int n  = lane & 15;
  const int hi = lane >> 4;
  const float bv0 = bo[d0 + n];
  const float bv1 = bo[d0 + 16 + n];
#pragma unroll
  for (int r = 0; r < 8; ++r) {
    const size_t r0 = (size_t)(m0 + r + 8 * hi) * D_MODEL;
    const size_t r1 = (size_t)(m0 + 16 + r + 8 * hi) * D_MODEL;
    out[r0 + d0 + n]      = c00[r] + bv0;
    out[r0 + d0 + 16 + n] = c01[r] + bv1;
    out[r1 + d0 + n]      = c10[r] + bv0;
    out[r1 + d0 + 16 + n] = c11[r] + bv1;
  }
}

// ---------------------------------------------------------------------------
extern "C" void kernel_launch(void* const* d_in, const int* in_sizes, int n_in,
                              void* d_out, int out_size, void* d_ws, size_t ws_size,
                              hipStream_t stream) {
  (void)in_sizes; (void)n_in; (void)out_size;
  const float* x   = (const float*)d_in[0];
  const float* Wq  = (const float*)d_in[1];
  const float* bq  = (const float*)d_in[2];
  const float* Wk  = (const float*)d_in[3];
  const float* bk  = (const float*)d_in[4];
  const float* Wov = (const float*)d_in[5];
  const float* bov = (const float*)d_in[6];
  const float* Wo  = (const float*)d_in[7];
  const float* bo  = (const float*)d_in[8];
  float* out = (float*)d_out;

  // Workspace layout (floats): Q,K,V,O (4 MB each) then partition partials.
  const size_t fMat = (size_t)TOTROWS * D_HID;       // 1,048,576 floats
  int P = 4;                                         // key partitions per q-tile
  {
    const size_t need = (4 * fMat + (size_t)NQT * P * (16 * D_HID + 32)) * sizeof(float);
    if (need > ws_size) P = 1;                       // deterministic host fallback
  }
  float* Qw   = (float*)d_ws;
  float* Kw   = Qw + fMat;
  float* Vw   = Kw + fMat;
  float* Ow   = Vw + fMat;
  float* Opar = Ow + fMat;
  float* Mpar = Opar + (size_t)NQT * P * 16 * D_HID;
  float* Lpar = Mpar + (size_t)NQT * P * 16;

  {  // QKV: 512 row tiles x 6 col-pair tiles = 3072 waves
    const int waves  = (TOTROWS / 32) * 6;
    const int tpb    = 256;
    const int blocks = (waves * 32 + tpb - 1) / tpb;
    qkv_kernel<<<blocks, tpb, 0, stream>>>(x, Wq, bq, Wk, bk, Wov, bov, Qw, Kw, Vw);
  }
  {  // attention partials: NQT * P waves, 4 waves/block
    const int waves  = NQT * P;
    const int blocks = (waves * 32 + 127) / 128;
    attn_kernel<<<blocks, 128, 0, stream>>>(Qw, Kw, Vw, Opar, Mpar, Lpar, P);
    combine_kernel<<<NQT, 256, 0, stream>>>(Opar, Mpar, Lpar, Ow, P);
  }
  {  // output projection: 512 row tiles x 32 col tiles = 16384 waves
    const int waves  = (TOTROWS / 32) * 32;
    const int tpb    = 256;
    const int blocks = (waves * 32 + tpb - 1) / tpb;
    proj_kernel<<<blocks, tpb, 0, stream>>>(Ow, Wo, bo, out);
  }
}